// DQSEncoder_45303315038580
// MI455X (gfx1250) — compile-verified
//
#include <hip/hip_runtime.h>
#include <hip/hip_bf16.h>

typedef __attribute__((ext_vector_type(2))) float v2f;
typedef __attribute__((ext_vector_type(4))) float v4f;
typedef __attribute__((ext_vector_type(8))) float v8f;

#define N_MODES 10

// One wave (32 lanes) handles a tile of 16 consecutive v's.
// GEMM: qs[v, c] = sum_{k=0..19} weights[v,k] * Bmat[k,c],  c = b*4 + f  (64 cols)
// Bmat rows k<10 come from x; rows k>=10 are the rest-pose constant [0,0,0,1].
// 5 K-steps x 4 N-tiles of V_WMMA_F32_16X16X4_F32, then VALU normalize/rotate.
__global__ __launch_bounds__(256) void DQSEncoder_kernel(
    const float* __restrict__ x,    // (16, 10, 4)
    const float* __restrict__ w,    // (V, 20)
    const float* __restrict__ vr,   // (V, 4)
    float* __restrict__ out,        // (16, V*4)
    int V)
{
    const int lane = threadIdx.x & 31;
    const int wave = (blockIdx.x << 3) + (threadIdx.x >> 5);
    const int v0   = wave << 4;
    if (v0 >= V) return;            // wave-uniform: EXEC stays all-1s for WMMA

    const int col  = lane & 15;     // N for B/C/D tiles; M-row for A tile
    const int half = lane >> 4;

    // Warm L2/L0 with this tile's VR rows while the GEMM runs.
    __builtin_prefetch(vr + (size_t)v0 * 4, 0, 3);

    // ---- Build B operand tiles (independent of v -> hoisted, 20 v2f regs) ----
    // B layout (4x16 f32, 2 VGPRs): lane holds column N=lane&15, K = 2*half + j.
    v2f Bt[4][5];
#pragma unroll
    for (int t = 0; t < 4; ++t) {
        const int c = t * 16 + col;
        const int b = c >> 2;
        const int f = c & 3;
#pragma unroll
        for (int ks = 0; ks < 5; ++ks) {
#pragma unroll
            for (int j = 0; j < 2; ++j) {
                const int k = ks * 4 + half * 2 + j;
                float val;
                if (k < N_MODES) val = x[b * (N_MODES * 4) + k * 4 + f];
                else             val = (f == 3) ? 1.0f : 0.0f;
                Bt[t][ks][j] = val;
            }
        }
    }

    // ---- GEMM over K=20: A = weights[v0+row, k..k+3] as 16x4 f32 tile ----
    // A layout: lane holds row M=lane&15, K = 2*half + j -> contiguous float2 load.
    v8f acc[4];
#pragma unroll
    for (int t = 0; t < 4; ++t) acc[t] = (v8f){0,0,0,0,0,0,0,0};

    const float* wrow = w + (size_t)(v0 + col) * 20 + half * 2;
#pragma unroll
    for (int ks = 0; ks < 5; ++ks) {
        const v2f A = *(const v2f*)(wrow + ks * 4);
#pragma unroll
        for (int t = 0; t < 4; ++t) {
            acc[t] = __builtin_amdgcn_wmma_f32_16x16x4_f32(
                /*neg_a=*/false, A, /*neg_b=*/false, Bt[t][ks],
                /*c_mod=*/(short)0, acc[t],
                /*reuse_a=*/false, /*reuse_b=*/false);
        }
    }

    // ---- Epilogue: normalize quats, rotate VR, store ----
    // C/D layout: acc[t][j] at lane L = qs[v0 + j + 8*(L>>4)][16t + (L&15)]
    const int f    = lane & 3;        // column % 4 == lane % 4
    const int base = lane & ~3;       // first lane of this 4-lane quat group
#pragma unroll
    for (int j = 0; j < 8; ++j) {
        const int vrow = v0 + j + half * 8;
        const v4f vv = *(const v4f*)(vr + (size_t)vrow * 4);
#pragma unroll
        for (int t = 0; t < 4; ++t) {
            float q = acc[t][j];
            // |q|^2 across the 4 lanes holding (x,y,z,w) of this (b,v)
            float s = q * q;
            s += __shfl_xor(s, 1);
            s += __shfl_xor(s, 2);
            q *= rsqrtf(s);
            // gather the full normalized quaternion into every lane
            const float qx = __shfl(q, base + 0);
            const float qy = __shfl(q, base + 1);
            const float qz = __shfl(q, base + 2);
            const float qw = __shfl(q, base + 3);

            const float xx = qx*qx, yy = qy*qy, zz = qz*qz;
            const float xy = qx*qy, xz = qx*qz, yz = qy*qz;
            const float wx = qw*qx, wy = qw*qy, wz = qw*qz;

            float r0, r1, r2;                      // rotation-matrix row i = f
            if (f == 0)      { r0 = 1.0f - 2.0f*(yy+zz); r1 = 2.0f*(xy-wz);        r2 = 2.0f*(xz+wy); }
            else if (f == 1) { r0 = 2.0f*(xy+wz);        r1 = 1.0f - 2.0f*(xx+zz); r2 = 2.0f*(yz-wx); }
            else             { r0 = 2.0f*(xz-wy);        r1 = 2.0f*(yz+wx);        r2 = 1.0f - 2.0f*(xx+yy); }

            float y = r0 * vv[0] + r1 * vv[1] + r2 * vv[2];
            if (f == 3) y = vv[3];                 // w passthrough

            const int b = (t * 16 + col) >> 2;
            out[(size_t)b * ((size_t)V * 4) + (size_t)vrow * 4 + f] = y;
        }
    }
}

extern "C" void kernel_launch(void* const* d_in, const int* in_sizes, int n_in,
                              void* d_out, int out_size, void* d_ws, size_t ws_size,
                              hipStream_t stream) {
    const float* x  = (const float*)d_in[0];   // (16, 10, 4)
    const float* w  = (const float*)d_in[1];   // (V, 20)
    const float* vr = (const float*)d_in[2];   // (V, 4)
    float* out = (float*)d_out;                // (16, V*4)

    const int V = in_sizes[2] / 4;             // 262144
    const int tiles = (V + 15) / 16;           // one 16-v tile per wave
    const int wavesPerBlock = 8;               // 256 threads, wave32
    const int blocks = (tiles + wavesPerBlock - 1) / wavesPerBlock;

    DQSEncoder_kernel<<<blocks, 256, 0, stream>>>(x, w, vr, out, V);
}